// GlobalAttn_4475355922476
// MI455X (gfx1250) — compile-verified
//
#include <hip/hip_runtime.h>
#include <hip/hip_bf16.h>
#include <math.h>

// Problem constants (from reference): L=2048, B=32, H=1024
#define LSEQ 2048
#define BATCH 32
#define HDIM 1024

typedef __attribute__((ext_vector_type(2))) float v2f;
typedef __attribute__((ext_vector_type(4))) float v4f;
typedef __attribute__((ext_vector_type(8))) float v8f;

// ---------------------------------------------------------------------------
// Kernel 1: v[b,h] = sum_k hidden[b,k] * W[k,h]   (V = Hs(32x1024) @ W(1024x1024))
// Implemented with V_WMMA_F32_16X16X4_F32. One wave per block computes a
// 32(M) x 16(N) slab: two 16x16 D tiles, K looped in steps of 4.
// ISA layouts (CDNA5 ISA §7.12.2):
//   A 16x4 f32 : lanes 0-15 -> {K=0,K=1}, lanes 16-31 -> {K=2,K=3}, M = lane&15
//   B 4x16 f32 : VGPR j, lanes 0-15 -> row K=j (j in {0,1}), lanes 16-31 -> K=j+2, N = lane&15
//   D 16x16 f32: VGPR j, lanes 0-15 -> M=j, lanes 16-31 -> M=j+8, N = lane&15
// ---------------------------------------------------------------------------
__global__ __launch_bounds__(32) void ga_proj_vec_wmma(
    const float* __restrict__ Hs,   // [BATCH][HDIM]
    const float* __restrict__ W,    // [HDIM][HDIM]  (W[k][h], row-major)
    float* __restrict__ Vv)         // [BATCH][HDIM]
{
    const int n0   = blockIdx.x * 16;      // N-tile start (64 blocks)
    const int lane = threadIdx.x;          // 0..31, wave32
    const int half = lane >> 4;            // 0: K pair {0,1}; 1: K pair {2,3}
    const int lid  = lane & 15;

    v8f acc0 = {};   // M-tile 0 (rows 0..15)
    v8f acc1 = {};   // M-tile 1 (rows 16..31)

    for (int k = 0; k < HDIM; k += 4) {
        const int ka = k + half * 2;
        v2f a0, a1, bb;
        a0.x = Hs[(lid)      * HDIM + ka];
        a0.y = Hs[(lid)      * HDIM + ka + 1];
        a1.x = Hs[(16 + lid) * HDIM + ka];
        a1.y = Hs[(16 + lid) * HDIM + ka + 1];
        bb.x = W[(size_t)(ka)     * HDIM + n0 + lid];
        bb.y = W[(size_t)(ka + 1) * HDIM + n0 + lid];
        acc0 = __builtin_amdgcn_wmma_f32_16x16x4_f32(false, a0, false, bb,
                                                     (short)0, acc0, false, false);
        acc1 = __builtin_amdgcn_wmma_f32_16x16x4_f32(false, a1, false, bb,
                                                     (short)0, acc1, false, false);
    }

    // Scatter D tiles: row m = (half*8 + j), col = n0 + lid
    #pragma unroll
    for (int j = 0; j < 8; ++j) {
        Vv[(size_t)(half * 8 + j)      * HDIM + n0 + lid] = acc0[j];
        Vv[(size_t)(16 + half * 8 + j) * HDIM + n0 + lid] = acc1[j];
    }
}

// ---------------------------------------------------------------------------
// Kernel 2 (bandwidth-critical): scores[b,l] = e[l,b,:] . v[b,:]
// Streams all 256 MB of encoder_outputs exactly once -> non-temporal (TH=NT)
// b128 loads so the zero-reuse stream doesn't evict the L2-resident v table.
// Block = 256 threads (8 waves) pinned to one b; v[b] staged in LDS (4 KB).
// Each wave computes 4 dot products, lane-cooperative, wave32 butterfly reduce.
// ---------------------------------------------------------------------------
#define LPW 4  // l's per wave
__global__ __launch_bounds__(256) void ga_scores(
    const float* __restrict__ E,    // [LSEQ][BATCH][HDIM]
    const float* __restrict__ Vv,   // [BATCH][HDIM]
    float* __restrict__ S)          // [BATCH][LSEQ]
{
    __shared__ __align__(16) float vsh[HDIM];
    const int b   = blockIdx.y;
    const int tid = threadIdx.x;

    // Stage v[b] into LDS: 256 threads x one 16B vector each = 4 KB
    ((v4f*)vsh)[tid] = ((const v4f*)(Vv + (size_t)b * HDIM))[tid];
    __syncthreads();

    const int wave = tid >> 5;
    const int lane = tid & 31;

    #pragma unroll
    for (int i = 0; i < LPW; ++i) {
        const int l = blockIdx.x * (8 * LPW) + wave * LPW + i;
        const v4f* e4 = (const v4f*)(E + ((size_t)l * BATCH + b) * HDIM);
        float sum = 0.f;
        #pragma unroll
        for (int c = 0; c < 8; ++c) {            // 8 * 32 lanes * 4 f32 = 1024 f32
            const v4f ev = __builtin_nontemporal_load(&e4[lane + c * 32]);
            const v4f vv = ((const v4f*)vsh)[lane + c * 32];
            sum += ev.x * vv.x + ev.y * vv.y + ev.z * vv.z + ev.w * vv.w;
        }
        #pragma unroll
        for (int m = 16; m >= 1; m >>= 1)
            sum += __shfl_xor(sum, m, 32);
        if (lane == 0)
            S[(size_t)b * LSEQ + l] = sum;
    }
}

// ---------------------------------------------------------------------------
// Kernel 3: row softmax over l (32 rows x 2048). One block per b.
// Note: bias term c[b] = hid[b].bias is constant per row -> softmax-invariant,
// so it is deliberately omitted.
// ---------------------------------------------------------------------------
__global__ __launch_bounds__(256) void ga_softmax(
    const float* __restrict__ S,    // [BATCH][LSEQ]
    float* __restrict__ O)          // [BATCH][LSEQ]
{
    const int b   = blockIdx.x;
    const int tid = threadIdx.x;
    __shared__ float red[8];

    float vals[8];
    float m = -INFINITY;
    #pragma unroll
    for (int i = 0; i < 8; ++i) {
        vals[i] = S[(size_t)b * LSEQ + tid + i * 256];
        m = fmaxf(m, vals[i]);
    }
    #pragma unroll
    for (int s = 16; s >= 1; s >>= 1)
        m = fmaxf(m, __shfl_xor(m, s, 32));
    if ((tid & 31) == 0) red[tid >> 5] = m;
    __syncthreads();
    float bm = red[0];
    #pragma unroll
    for (int w = 1; w < 8; ++w) bm = fmaxf(bm, red[w]);
    __syncthreads();

    float sum = 0.f;
    #pragma unroll
    for (int i = 0; i < 8; ++i) {
        vals[i] = __expf(vals[i] - bm);
        sum += vals[i];
    }
    #pragma unroll
    for (int s = 16; s >= 1; s >>= 1)
        sum += __shfl_xor(sum, s, 32);
    if ((tid & 31) == 0) red[tid >> 5] = sum;
    __syncthreads();
    float bs = 0.f;
    #pragma unroll
    for (int w = 0; w < 8; ++w) bs += red[w];
    const float inv = 1.0f / bs;
    #pragma unroll
    for (int i = 0; i < 8; ++i)
        O[(size_t)b * LSEQ + tid + i * 256] = vals[i] * inv;
}

// ---------------------------------------------------------------------------
// Launch: inputs in setup_inputs() order:
//   d_in[0] hidden_state (B*H f32), d_in[1] encoder_outputs (L*B*H f32),
//   d_in[2] W (H*H f32), d_in[3] b (H f32, unused — softmax invariant).
// d_out: (B, L) f32 softmax. d_ws: [v : B*H f32][scores : B*L f32]
// ---------------------------------------------------------------------------
extern "C" void kernel_launch(void* const* d_in, const int* in_sizes, int n_in,
                              void* d_out, int out_size, void* d_ws, size_t ws_size,
                              hipStream_t stream) {
    const float* hid = (const float*)d_in[0];
    const float* enc = (const float*)d_in[1];
    const float* W   = (const float*)d_in[2];
    (void)in_sizes; (void)n_in; (void)out_size; (void)ws_size;

    float* v      = (float*)d_ws;                       // B*H = 32768 floats
    float* scores = v + (size_t)BATCH * HDIM;           // B*L = 65536 floats
    float* out    = (float*)d_out;

    ga_proj_vec_wmma<<<HDIM / 16, 32, 0, stream>>>(hid, W, v);
    ga_scores<<<dim3(LSEQ / (8 * LPW), BATCH), 256, 0, stream>>>(enc, v, scores);
    ga_softmax<<<BATCH, 256, 0, stream>>>(scores, out);
}